// MDL_X1_23459111371234
// MI455X (gfx1250) — compile-verified
//
#include <hip/hip_runtime.h>
#include <hip/hip_bf16.h>
#include <math.h>

#define DFEAT 128
#define TPITCH 132   // LDS pitch in floats (128 + 4 pad -> conflict-free column access)

typedef __attribute__((ext_vector_type(2))) float v2f;
typedef __attribute__((ext_vector_type(8))) float v8f;

// ---------------------------------------------------------------------------
// zero workspace
// ---------------------------------------------------------------------------
__global__ void MDL_zero_kernel(float4* __restrict__ p, long n4) {
    long i = (long)blockIdx.x * blockDim.x + threadIdx.x;
    if (i < n4) p[i] = make_float4(0.f, 0.f, 0.f, 0.f);
}

// ---------------------------------------------------------------------------
// SpMM: y[r] += v * x[c]  — one wave32 per edge, lane = one float4 of the row
// (coalesced 512B gather of x[c], atomic f32 scatter into y[r])
// ---------------------------------------------------------------------------
__global__ __launch_bounds__(256)
void MDL_spmm_kernel(const int* __restrict__ rows, const int* __restrict__ cols,
                     const float* __restrict__ vals, const float* __restrict__ x,
                     float* __restrict__ y, int nedges) {
    int wave = (int)((blockIdx.x * (unsigned)blockDim.x + threadIdx.x) >> 5);
    int lane = threadIdx.x & 31;
    if (wave >= nedges) return;
    int r = rows[wave];
    int c = cols[wave];
    float v = vals[wave];
    const float4* xs = (const float4*)(x + (size_t)c * DFEAT);
    float4 xv = xs[lane];
    float* yd = y + (size_t)r * DFEAT + lane * 4;
    atomicAdd(yd + 0, v * xv.x);
    atomicAdd(yd + 1, v * xv.y);
    atomicAdd(yd + 2, v * xv.z);
    atomicAdd(yd + 3, v * xv.w);
}

// ---------------------------------------------------------------------------
// 16x16 output tile of  out = A(16x128, LDS) * W^T(j0..j0+15 rows) + bias
// via 32x V_WMMA_F32_16X16X4_F32.
// A operand: lane l holds M=l&15, K-half = (l>>4)*2  (ISA 7.12.2, 32-bit A 16x4)
// B operand: lane l holds N=l&15, same K-half; B[k][n] = W[(j0+n)*128 + k]
// C/D: VGPR v -> row v + 8*(l>=16), col = l&15
// ---------------------------------------------------------------------------
__device__ __forceinline__ v8f MDL_gemm16(const float* __restrict__ A,
                                          const float* __restrict__ W,
                                          const float* __restrict__ bias, int j0) {
    const int lane = threadIdx.x & 31;
    const int n    = lane & 15;
    const int kh   = (lane >> 4) << 1;        // 0 or 2
    const float bv = bias[j0 + n];
    v8f acc = {bv, bv, bv, bv, bv, bv, bv, bv};
    const float* arow = A + n * TPITCH;                    // M == (lane&15) for A operand
    const float* wrow = W + (size_t)(j0 + n) * DFEAT;      // row of W supplies B column n
#pragma unroll
    for (int k0 = 0; k0 < DFEAT; k0 += 4) {
        v2f av, bvv;
        av.x  = arow[k0 + kh];
        av.y  = arow[k0 + kh + 1];
        bvv.x = wrow[k0 + kh];
        bvv.y = wrow[k0 + kh + 1];
        acc = __builtin_amdgcn_wmma_f32_16x16x4_f32(
            /*neg_a=*/false, av, /*neg_b=*/false, bvv,
            /*c_mod=*/(short)0, acc, /*reuse_a=*/false, /*reuse_b=*/false);
    }
    return acc;
}

__device__ __forceinline__ void MDL_tile_to_lds(float* __restrict__ dst, int j0, v8f acc) {
    const int lane = threadIdx.x & 31;
    const int n  = lane & 15;
    const int mb = (lane >> 4) << 3;
#pragma unroll
    for (int v = 0; v < 8; ++v)
        dst[(mb + v) * TPITCH + j0 + n] = acc[v];
}

__device__ __forceinline__ void MDL_tile_to_global(float* __restrict__ dst, int r0, int j0,
                                                   v8f acc, int nrows) {
    const int lane = threadIdx.x & 31;
    const int n  = lane & 15;
    const int mb = (lane >> 4) << 3;
#pragma unroll
    for (int v = 0; v < 8; ++v) {
        int row = r0 + mb + v;
        if (row < nrows) dst[(size_t)row * DFEAT + j0 + n] = acc[v];
    }
}

// ---------------------------------------------------------------------------
// Fused head: per block of 16 rows of gnn, compute
//   emb = L2norm(gnn @ embW^T + embB)
//   lft = tanh( (tanh(gnn@W0^T+b0)+gnn) @ W1^T + b1 )   (same for rgt)
// 8 waves/block, one 16x16 WMMA tile per wave per GEMM.
// ---------------------------------------------------------------------------
__global__ __launch_bounds__(256)
void MDL_head_kernel(const float* __restrict__ gnn, int nrows,
                     const float* __restrict__ embW, const float* __restrict__ embB,
                     const float* __restrict__ lftW, const float* __restrict__ lftB,
                     const float* __restrict__ rgtW, const float* __restrict__ rgtB,
                     float* __restrict__ out_emb, float* __restrict__ out_lft,
                     float* __restrict__ out_rgt) {
    __shared__ float Atile[16 * TPITCH];   // gnn row tile (kept alive for residuals)
    __shared__ float Btile[16 * TPITCH];   // staging for intermediates
    __shared__ float norms[16];

    const int tid  = threadIdx.x;
    const int lane = tid & 31;
    const int j0   = (tid >> 5) << 4;      // this wave's 16-column slice
    const int r0   = blockIdx.x << 4;

    // ---- load 16x128 gnn tile into LDS (float4, fully coalesced) ----
#pragma unroll
    for (int t = 0; t < 2; ++t) {
        int idx = tid + t * 256;           // float4 index 0..511 (32 per row)
        int row = idx >> 5;
        int c4  = (idx & 31) << 2;
        float4 val = make_float4(0.f, 0.f, 0.f, 0.f);
        if (r0 + row < nrows)
            val = ((const float4*)(gnn + (size_t)(r0 + row) * DFEAT))[idx & 31];
        float* d = Atile + row * TPITCH + c4;
        d[0] = val.x; d[1] = val.y; d[2] = val.z; d[3] = val.w;
    }
    __syncthreads();

    // ---- embedding head: GEMM -> LDS -> row L2 norm -> store ----
    {
        v8f e = MDL_gemm16(Atile, embW, embB, j0);
        MDL_tile_to_lds(Btile, j0, e);
    }
    __syncthreads();
    if (tid < 16) {
        float s = 0.f;
        const float* row = Btile + tid * TPITCH;
#pragma unroll
        for (int c = 0; c < DFEAT; ++c) s += row[c] * row[c];
        norms[tid] = fmaxf(sqrtf(s), 1e-12f);
    }
    __syncthreads();
#pragma unroll
    for (int t = 0; t < 8; ++t) {
        int idx = tid + t * 256;           // 0..2047
        int row = idx >> 7, col = idx & 127;
        if (r0 + row < nrows)
            out_emb[(size_t)(r0 + row) * DFEAT + col] = Btile[row * TPITCH + col] / norms[row];
    }
    __syncthreads();

    // ---- left tower ----
    {
        v8f h = MDL_gemm16(Atile, lftW, lftB, j0);
        const int n = lane & 15, mb = (lane >> 4) << 3;
#pragma unroll
        for (int v = 0; v < 8; ++v)
            h[v] = tanhf(h[v]) + Atile[(mb + v) * TPITCH + j0 + n];   // residual (layer 0)
        MDL_tile_to_lds(Btile, j0, h);
    }
    __syncthreads();
    {
        v8f h = MDL_gemm16(Btile, lftW + DFEAT * DFEAT, lftB + DFEAT, j0);
#pragma unroll
        for (int v = 0; v < 8; ++v) h[v] = tanhf(h[v]);
        MDL_tile_to_global(out_lft, r0, j0, h, nrows);
    }
    __syncthreads();

    // ---- right tower ----
    {
        v8f h = MDL_gemm16(Atile, rgtW, rgtB, j0);
        const int n = lane & 15, mb = (lane >> 4) << 3;
#pragma unroll
        for (int v = 0; v < 8; ++v)
            h[v] = tanhf(h[v]) + Atile[(mb + v) * TPITCH + j0 + n];
        MDL_tile_to_lds(Btile, j0, h);
    }
    __syncthreads();
    {
        v8f h = MDL_gemm16(Btile, rgtW + DFEAT * DFEAT, rgtB + DFEAT, j0);
#pragma unroll
        for (int v = 0; v < 8; ++v) h[v] = tanhf(h[v]);
        MDL_tile_to_global(out_rgt, r0, j0, h, nrows);
    }
}

// ---------------------------------------------------------------------------
extern "C" void kernel_launch(void* const* d_in, const int* in_sizes, int n_in,
                              void* d_out, int out_size, void* d_ws, size_t ws_size,
                              hipStream_t stream) {
    const float* feat  = (const float*)d_in[0];
    const int*   srows = (const int*)  d_in[1];
    const int*   scols = (const int*)  d_in[2];
    const float* svals = (const float*)d_in[3];
    // d_in[4], d_in[5]: feat_W / feat_b — dead code in the reference (not returned)
    const float* embW  = (const float*)d_in[6];
    const float* embB  = (const float*)d_in[7];
    const float* lftW  = (const float*)d_in[8];
    const float* lftB  = (const float*)d_in[9];
    const float* rgtW  = (const float*)d_in[10];
    const float* rgtB  = (const float*)d_in[11];

    const int nnodes = in_sizes[0] / DFEAT;
    const int nedges = in_sizes[1];

    float* gnn1 = (float*)d_ws;                       // A @ feat
    float* gnn2 = gnn1 + (size_t)nnodes * DFEAT;      // A @ (A @ feat)
    float* out_emb = (float*)d_out;
    float* out_lft = out_emb + (size_t)nnodes * DFEAT;
    float* out_rgt = out_lft + (size_t)nnodes * DFEAT;

    // zero both accumulation buffers (2*N*D floats, contiguous)
    long n4 = ((long)nnodes * DFEAT * 2) / 4;
    MDL_zero_kernel<<<(int)((n4 + 255) / 256), 256, 0, stream>>>((float4*)gnn1, n4);

    // two sparse aggregation layers (wave per edge, 8 waves per block)
    int spmm_blocks = (nedges + 7) / 8;
    MDL_spmm_kernel<<<spmm_blocks, 256, 0, stream>>>(srows, scols, svals, feat, gnn1, nedges);
    MDL_spmm_kernel<<<spmm_blocks, 256, 0, stream>>>(srows, scols, svals, gnn1, gnn2, nedges);

    // fused WMMA dense heads
    int head_blocks = (nnodes + 15) / 16;
    MDL_head_kernel<<<head_blocks, 256, 0, stream>>>(gnn2, nnodes,
                                                     embW, embB, lftW, lftB, rgtW, rgtB,
                                                     out_emb, out_lft, out_rgt);
}